// CausalPredictor_46462956208724
// MI455X (gfx1250) — compile-verified
//
#include <hip/hip_runtime.h>

// Reference collapses: softmax over a size-1 axis == 1.0, so
//   out[l, c] = prior[0] * dic[1, 0, c]   (broadcast of one scaled 4KB row)
// Pure HBM-store-bound: 512 MiB writes -> ~22us floor at 23.3 TB/s.
//
// Strategy: stage scaled row in LDS once per block, then stream it to global
// memory with gfx1250 async LDS->global stores (global_store_async_from_lds_b128),
// non-temporal hint, ASYNCcnt-pipelined.

#define C_DIM 1024            // channels (dic is [2,1,C])
#define THREADS (C_DIM / 4)   // 256 threads: one float4 of the row per thread
#define ROWS_PER_BLOCK 64     // 131072 / 64 = 2048 blocks

__global__ __launch_bounds__(THREADS)
void CausalPredictor_bcast_kernel(const float* __restrict__ dic,
                                  const float* __restrict__ prior,
                                  float* __restrict__ out,
                                  int rows_per_block,
                                  long total_rows)
{
    __shared__ float row[C_DIM];

    const int t = threadIdx.x;            // 0..255, lane-slice of the row
    const float p = prior[0];

    // dic_z = dic[1:].reshape(1,C)  -> flat offset C_DIM
    float4 v = ((const float4*)(dic + C_DIM))[t];
    v.x *= p; v.y *= p; v.z *= p; v.w *= p;
    ((float4*)row)[t] = v;
    __syncthreads();                      // LDS row visible to async engine

    // 32-bit LDS byte offset of this thread's 16B slice. Generic LDS pointers
    // carry the LDS offset in addr[31:0] (aperture in the high bits), so the
    // truncation yields the DS-space address the async op wants.
    const unsigned lds_off = (unsigned)(size_t)row + (unsigned)(t * 16);

    long row0 = (long)blockIdx.x * rows_per_block;
    float* gp = out + row0 * C_DIM + t * 4;

    int nrows = rows_per_block;
    if (row0 + nrows > total_rows) nrows = (int)(total_rows - row0);

    // One async b128 store per lane per row: a wave32 emits one instruction
    // moving 512 contiguous bytes LDS->HBM, no data VGPRs, pipelined on ASYNCcnt.
    for (int r = 0; r < nrows; ++r) {
        asm volatile("global_store_async_from_lds_b128 %0, %1, off th:TH_STORE_NT"
                     :
                     : "v"(gp), "v"(lds_off)
                     : "memory");
        gp += C_DIM;
    }

    // Retire all outstanding async LDS->global stores before wave exit.
    asm volatile("s_wait_asynccnt 0x0" ::: "memory");
}

extern "C" void kernel_launch(void* const* d_in, const int* in_sizes, int n_in,
                              void* d_out, int out_size, void* d_ws, size_t ws_size,
                              hipStream_t stream) {
    // setup_inputs order: x(0), xm(1), Wy_w(2), Wy_b(3), Wz_w(4), Wz_b(5), dic(6), prior(7)
    const float* dic   = (const float*)d_in[6];   // [2,1,C]
    const float* prior = (const float*)d_in[7];   // [1]
    float* out = (float*)d_out;                   // [L,1,C] flat

    const long total_rows = (long)out_size / C_DIM;   // L = 131072
    const int blocks = (int)((total_rows + ROWS_PER_BLOCK - 1) / ROWS_PER_BLOCK);

    CausalPredictor_bcast_kernel<<<blocks, THREADS, 0, stream>>>(
        dic, prior, out, ROWS_PER_BLOCK, total_rows);
}